// Conv1DTokenEncoder_51608327028996
// MI455X (gfx1250) — compile-verified
//
#include <hip/hip_runtime.h>
#include <hip/hip_bf16.h>

// Problem sizes (fixed by the reference)
#define BB 16
#define TT 2048
#define LL 1024
#define DD 512
#define KK 5
#define NROWS (BB * TT)      // 32768 token rows
#define ROWS_PER_BLOCK 16
#define THREADS 512          // 16 wave32 waves

typedef __attribute__((ext_vector_type(2))) float v2f;
typedef __attribute__((ext_vector_type(4))) float v4f;
typedef __attribute__((ext_vector_type(8))) float v8f;

__device__ __forceinline__ float wave_reduce_sum(float v) {
#pragma unroll
  for (int off = 16; off > 0; off >>= 1)
    v += __shfl_xor(v, off, 32);
  return v;
}

__global__ __launch_bounds__(THREADS)
void conv1d_token_encoder_kernel(const float* __restrict__ x,
                                 const float* __restrict__ w,
                                 const float* __restrict__ bias,
                                 float* __restrict__ out) {
  __shared__ float sS[ROWS_PER_BLOCK * 8];   // padded S (K=8) per row, scaled by 1/L

  const int lane = threadIdx.x & 31;
  const int wave = threadIdx.x >> 5;         // 0..15
  const int row  = blockIdx.x * ROWS_PER_BLOCK + wave;

  // ---------------- Phase 1: row reduction (streaming, NT b128 loads) ----------------
  {
    const v4f* xr = (const v4f*)(x + (size_t)row * LL);
    float psum = 0.0f;
    v4f first = {0.f, 0.f, 0.f, 0.f};
    v4f last  = {0.f, 0.f, 0.f, 0.f};
#pragma unroll
    for (int c = 0; c < 8; ++c) {
      v4f v = __builtin_nontemporal_load(&xr[c * 32 + lane]);  // idx c*128 + lane*4 .. +3
      if (c == 0) first = v;
      if (c == 7) last = v;
      psum += (v[0] + v[1]) + (v[2] + v[3]);
    }
    const float total = wave_reduce_sum(psum);
    // edges: x[0],x[1] live in lane 0's first vec; x[L-2],x[L-1] in lane 31's last vec
    const float x0  = __shfl(first[0], 0, 32);
    const float x1  = __shfl(first[1], 0, 32);
    const float xm2 = __shfl(last[2], 31, 32);
    const float xm1 = __shfl(last[3], 31, 32);

    if (lane < 8) {
      float s;
      switch (lane) {
        case 0: s = total - xm2 - xm1; break;
        case 1: s = total - xm1;       break;
        case 2: s = total;             break;
        case 3: s = total - x0;        break;
        case 4: s = total - x0 - x1;   break;
        default: s = 0.0f;             break;    // K padded 5 -> 8
      }
      sS[wave * 8 + lane] = s * (1.0f / (float)LL);
    }
  }
  __syncthreads();

  // ---------------- Phase 2: [16 x 8] x [8 x 512] via V_WMMA_F32_16X16X4_F32 ----------------
  const int m = lane & 15;      // output row within tile / B column index
  const int h = lane >> 4;      // lane half selects K pair
  const int n = m;              // output column within d-tile

  // A fragments (identical for every d-tile): lane holds K = 2h,2h+1 (+4 for chunk 1)
  const v2f a0 = *(const v2f*)&sS[m * 8 + 2 * h];
  const v2f a1 = *(const v2f*)&sS[m * 8 + 4 + 2 * h];

#pragma unroll
  for (int t = 0; t < 2; ++t) {
    const int d = (wave * 2 + t) * 16 + n;          // this lane's output channel
    const float* wd = w + (size_t)d * KK;           // w[d, :], K=5 row-major

    v2f b0, b1;
    b0.x = wd[2 * h];                               // K = 0/2
    b0.y = wd[2 * h + 1];                           // K = 1/3
    b1.x = (h == 0) ? wd[4] : 0.0f;                 // K = 4 valid, 6 padded
    b1.y = 0.0f;                                    // K = 5/7 padded

    const float bv = bias[d];
    v8f c;
#pragma unroll
    for (int r = 0; r < 8; ++r) c[r] = bv;          // bias broadcast into accumulator

    // D = A*B + C  (fp32 matrix pipe, two K=4 chunks)
    c = __builtin_amdgcn_wmma_f32_16x16x4_f32(false, a0, false, b0, (short)0, c, false, false);
    c = __builtin_amdgcn_wmma_f32_16x16x4_f32(false, a1, false, b1, (short)0, c, false, false);

    // C layout: VGPR r -> (row = r + 8h, col = n)
    float* obase = out + ((size_t)(blockIdx.x * ROWS_PER_BLOCK + 8 * h)) * DD + d;
#pragma unroll
    for (int r = 0; r < 8; ++r)
      __builtin_nontemporal_store(c[r], &obase[(size_t)r * DD]);
  }
}

extern "C" void kernel_launch(void* const* d_in, const int* in_sizes, int n_in,
                              void* d_out, int out_size, void* d_ws, size_t ws_size,
                              hipStream_t stream) {
  const float* x    = (const float*)d_in[0];   // [B, T, L]
  const float* w    = (const float*)d_in[1];   // [D, K]
  const float* bias = (const float*)d_in[2];   // [D]
  float* out        = (float*)d_out;           // [B, T, D]

  dim3 grid(NROWS / ROWS_PER_BLOCK);           // 2048 blocks
  dim3 block(THREADS);                         // 16 waves of 32
  conv1d_token_encoder_kernel<<<grid, block, 0, stream>>>(x, w, bias, out);
}